// IoU_2284922601770
// MI455X (gfx1250) — compile-verified
//
#include <hip/hip_runtime.h>

// Problem constants (match the reference).
#define KCLS   21
#define BB     16
#define HH     512
#define WW     512
#define HWPIX  (HH * WW)          // 262144 pixels per batch image
#define THREADS 128
#define PIXPT   4                 // pixels per thread (float4)
#define PIXPB   (THREADS * PIXPT) // 512 pixels per block
#define NCELL   (BB * KCLS)       // 336 (b,k) counter cells per histogram

// ---------------------------------------------------------------------------
// Kernel 0: zero the 3 * B * K uint32 counters in workspace (ws is poisoned
// by the harness, so this must run every call).
// ---------------------------------------------------------------------------
__global__ __launch_bounds__(1024) void iou_zero_kernel(unsigned* __restrict__ c) {
    int i = threadIdx.x;
    if (i < 3 * NCELL) c[i] = 0u;
}

// ---------------------------------------------------------------------------
// Kernel 1: streaming argmax + per-(b,k) histograms.
//   counters layout: [0, NCELL)         cp = |pred == k|
//                    [NCELL, 2*NCELL)   cs = |gt   == k|  (valid labels only)
//                    [2*NCELL, 3*NCELL) ci = |pred == gt == k|
// Each block handles PIXPB consecutive pixels of one batch image. The 21
// class planes are staged HBM -> LDS with async b128 copies (ASYNCcnt), and
// the argmax consumes row k after a staged wait ASYNCcnt <= 20-k so compute
// overlaps the in-flight tail of the async stream (async loads complete in
// issue order per the ISA).
// ---------------------------------------------------------------------------
__global__ __launch_bounds__(THREADS) void iou_hist_kernel(
    const float* __restrict__ pi,       // [B, K, H, W]
    const float* __restrict__ pistar,   // [B, H, W]
    unsigned*    __restrict__ counters)
{
    __shared__ float    tile[KCLS * PIXPB];  // 21 * 512 floats = 42 KB
    __shared__ unsigned hist[3 * KCLS];      // per-block cp/cs/ci bins

    const int tid    = threadIdx.x;
    const int b      = blockIdx.y;
    const int pixel0 = blockIdx.x * PIXPB;
    const int base   = tid * PIXPT;          // this thread's 4-pixel slot

    if (tid < 3 * KCLS) hist[tid] = 0u;

    // ---- Stage: issue 21 async global->LDS b128 copies (one per class). ----
    // Each lane moves its own 16 bytes; VDST VGPR carries the LDS byte offset
    // (low 32 bits of the generic pointer to __shared__), VADDR the 64-bit
    // global address. GV mode (no SADDR).
#pragma unroll
    for (int k = 0; k < KCLS; ++k) {
        const float* gp =
            pi + ((size_t)(b * KCLS + k)) * HWPIX + (size_t)pixel0 + base;
        unsigned lds_off =
            (unsigned)(unsigned long long)(const void*)&tile[k * PIXPB + base];
        asm volatile("global_load_async_to_lds_b128 %0, %1, off"
                     :: "v"(lds_off), "v"(gp)
                     : "memory");
    }

    // Start the label load now so it is in flight under the argmax.
    float4 s4 = *reinterpret_cast<const float4*>(
        &pistar[(size_t)b * HWPIX + (size_t)pixel0 + base]);

    __syncthreads();  // hist zeroing visible before the atomic phase

    // ---- Argmax over K for 4 pixels (first-max-wins, like jnp.argmax). ----
    // Each lane consumes only LDS bytes it loaded itself, and async loads
    // retire in order: before row k, wait ASYNCcnt <= (KCLS-1) - k.
    asm volatile("s_wait_asynccnt %0" :: "i"(KCLS - 1) : "memory");
    float4 v = *reinterpret_cast<const float4*>(&tile[base]);
    float b0 = v.x, b1 = v.y, b2 = v.z, b3 = v.w;
    int   i0 = 0,  i1 = 0,  i2 = 0,  i3 = 0;
#pragma unroll
    for (int k = 1; k < KCLS; ++k) {
        asm volatile("s_wait_asynccnt %0" :: "i"(KCLS - 1 - k) : "memory");
        float4 c = *reinterpret_cast<const float4*>(&tile[k * PIXPB + base]);
        if (c.x > b0) { b0 = c.x; i0 = k; }
        if (c.y > b1) { b1 = c.y; i1 = k; }
        if (c.z > b2) { b2 = c.z; i2 = k; }
        if (c.w > b3) { b3 = c.w; i3 = k; }
    }

    // ---- Labels: star = pi_star + 1 must be an exact integer in [0, K). ----
    {
        float sv[4] = { s4.x, s4.y, s4.z, s4.w };
        int   pr[4] = { i0, i1, i2, i3 };
#pragma unroll
        for (int j = 0; j < 4; ++j) {
            atomicAdd(&hist[pr[j]], 1u);                       // cp
            float s  = sv[j] + 1.0f;
            int   si = (int)s;
            bool valid = (s == (float)si) && (si >= 0) && (si < KCLS);
            if (valid) {
                atomicAdd(&hist[KCLS + si], 1u);               // cs
                if (si == pr[j]) atomicAdd(&hist[2 * KCLS + pr[j]], 1u); // ci
            }
        }
    }

    __syncthreads();

    // ---- Flush the block-local bins to the global per-(b,k) counters. ----
    if (tid < 3 * KCLS) {
        int which = tid / KCLS;          // 0=cp, 1=cs, 2=ci
        int cls   = tid % KCLS;
        atomicAdd(&counters[which * NCELL + b * KCLS + cls], hist[tid]);
    }
}

// ---------------------------------------------------------------------------
// Kernel 2: final reduction -> scalar mean IoU.
// ---------------------------------------------------------------------------
__global__ __launch_bounds__(512) void iou_final_kernel(
    const unsigned* __restrict__ counters, float* __restrict__ out)
{
    __shared__ float red[512];
    int i = threadIdx.x;
    float v = 0.0f;
    if (i < NCELL) {
        float cp = (float)counters[i];
        float cs = (float)counters[NCELL + i];
        float ci = (float)counters[2 * NCELL + i];
        v = ci / (cp + cs - ci + 1e-6f);
    }
    red[i] = v;
    __syncthreads();
#pragma unroll
    for (int s = 256; s > 0; s >>= 1) {
        if (i < s) red[i] += red[i + s];
        __syncthreads();
    }
    if (i == 0) out[0] = red[0] / (float)NCELL;
}

// ---------------------------------------------------------------------------
// Entry point.
// ---------------------------------------------------------------------------
extern "C" void kernel_launch(void* const* d_in, const int* in_sizes, int n_in,
                              void* d_out, int out_size, void* d_ws, size_t ws_size,
                              hipStream_t stream) {
    (void)in_sizes; (void)n_in; (void)out_size; (void)ws_size;

    const float* pi     = (const float*)d_in[0];  // [B, K, H, W] f32
    const float* pistar = (const float*)d_in[1];  // [B, H, W]    f32
    unsigned*    cnt    = (unsigned*)d_ws;        // 3 * B * K uint32
    float*       out    = (float*)d_out;          // scalar f32

    iou_zero_kernel<<<1, 1024, 0, stream>>>(cnt);

    dim3 grid(HWPIX / PIXPB, BB);  // (512, 16)
    iou_hist_kernel<<<grid, THREADS, 0, stream>>>(pi, pistar, cnt);

    iou_final_kernel<<<1, 512, 0, stream>>>(cnt, out);
}